// GCN_3layer_biased_67972152427189
// MI455X (gfx1250) — compile-verified
//
#include <hip/hip_runtime.h>
#include <math.h>

// ---------------------------------------------------------------------------
// GCN 3-layer (PyG GCNConv semantics) for MI455X / gfx1250, wave32, WMMA fp32.
//   N_NODES=100000, N_EDGES=800000, N_GRAPHS=256, IN_C=HID=96, OUT_C=16
// ---------------------------------------------------------------------------

#define N_NODES  100000
#define N_EDGES  800000
#define N_GRAPHS 256
#define HIDC     96
#define OUTC     16

typedef __attribute__((ext_vector_type(2))) float v2f;
typedef __attribute__((ext_vector_type(8))) float v8f;

// ---------------------------------------------------------------------------
// Degree: deg[dst] += 1 per edge (self-loop +1 folded into rsqrt kernel).
// ---------------------------------------------------------------------------
__global__ void gcn_degree(const long long* __restrict__ ei, float* __restrict__ deg,
                           int nEdges) {
    int e = blockIdx.x * blockDim.x + threadIdx.x;
    if (e >= nEdges) return;
    long long d = ei[(long long)nEdges + e];
    atomicAdd(&deg[d], 1.0f);
}

// dinv[i] = rsqrt(deg[i] + 1)   (in place on the deg buffer)
__global__ void gcn_rsqrt(float* __restrict__ deg, int n) {
    int i = blockIdx.x * blockDim.x + threadIdx.x;
    if (i >= n) return;
    deg[i] = rsqrtf(deg[i] + 1.0f);
}

// ---------------------------------------------------------------------------
// Dense GEMM  H[n,96] = A[n,96] @ W[96,96]  via V_WMMA_F32_16X16X4_F32.
// Block = 256 threads = 8 waves; wave w computes rows [blk*128 + 16w, +16).
// Each wave holds 6 accumulators (16x16 tiles covering all 96 columns).
//
// W is staged in LDS PRE-SWIZZLED into per-lane B fragments:
//   sW2[((k>>1)+half)*96 + n] = (W[k0][n], W[k0+1][n]),  k0 = (k&~3) + 2*half
// so each lane's B fragment is ONE aligned 8-byte ds_load_b64 — no VGPR
// repacking moves in the inner loop; the 6 column tiles are +128B immediates.
//
// Safe to run in place (out == in): a block writes only rows it alone reads.
// ---------------------------------------------------------------------------
__global__ __launch_bounds__(256) void gcn_gemm_wmma(
    const float* __restrict__ A, const float* __restrict__ W,
    float* __restrict__ H, int nrows) {
    __shared__ v2f sW2[(HIDC / 2) * HIDC];   // 48*96 float2 = 36 KB

    // Cooperative swizzled stage of W (one-time, L2-hot).
    // Entry i: kh = i/96 in [0,48), n = i%96; kchunk = kh>>1, half = kh&1;
    //          k0 = kchunk*4 + half*2; value = (W[k0][n], W[k0+1][n]).
    for (int i = threadIdx.x; i < (HIDC / 2) * HIDC; i += 256) {
        int kh = i / HIDC;
        int n  = i % HIDC;
        int k0 = (kh >> 1) * 4 + (kh & 1) * 2;
        v2f w;
        w.x = W[k0 * HIDC + n];
        w.y = W[(k0 + 1) * HIDC + n];
        sW2[i] = w;
    }
    __syncthreads();

    const int lane = threadIdx.x & 31;
    const int wave = threadIdx.x >> 5;
    const int half = lane >> 4;     // 0: lanes 0-15, 1: lanes 16-31
    const int l16  = lane & 15;
    const long long rowBase = (long long)blockIdx.x * 128 + (long long)wave * 16;

    v8f acc[6] = {};

    // A fragment source row for this lane (clamped; no branch => EXEC stays full)
    long long arow = rowBase + l16;
    if (arow > nrows - 1) arow = nrows - 1;
    const float* aptr = A + arow * HIDC;

    for (int kb = 0; kb < HIDC; kb += 4) {
        // A 16x4 f32 layout: v0 = K = kb+2*half, v1 = K+1  (ISA 7.12.2)
        // 8-byte aligned: row*96, kb, 2*half all even.
        v2f a = *(const v2f*)(aptr + kb + 2 * half);
        const v2f* brow = &sW2[((kb >> 1) + half) * HIDC + l16];
#pragma unroll
        for (int t = 0; t < 6; ++t) {
            v2f b = brow[t * 16];   // single ds_load_b64, +128B immediates
            acc[t] = __builtin_amdgcn_wmma_f32_16x16x4_f32(
                false, a, false, b, (short)0, acc[t], false, false);
        }
    }

    // C/D layout: VGPR vr -> M = vr + 8*half, N = l16  (ISA 7.12.2)
#pragma unroll
    for (int t = 0; t < 6; ++t) {
        const int n = t * 16 + l16;
#pragma unroll
        for (int vr = 0; vr < 8; ++vr) {
            long long row = rowBase + vr + 8 * half;
            if (row < nrows) H[row * HIDC + n] = acc[t][vr];
        }
    }
}

// ---------------------------------------------------------------------------
// Edge scatter: agg[dst] += h[src] * dinv[src]*dinv[dst]
// One thread per (edge, 4-channel group): 24 groups x 800K edges.
// h (38.4 MB) is L2-resident -> gathers & atomics are L2-bound, not HBM.
// ---------------------------------------------------------------------------
__global__ void gcn_edge_scatter(const long long* __restrict__ ei,
                                 const float* __restrict__ dinv,
                                 const float* __restrict__ h,
                                 float* __restrict__ agg, int nEdges) {
    long long tid = (long long)blockIdx.x * blockDim.x + threadIdx.x;
    long long e = tid / 24;
    int g = (int)(tid % 24);
    if (e >= nEdges) return;
    long long s = ei[e];
    long long d = ei[(long long)nEdges + e];
    float coef = dinv[s] * dinv[d];
    float4 v = *((const float4*)(h + s * HIDC) + g);
    float* ap = agg + d * HIDC + (long long)g * 4;
    atomicAdd(ap + 0, v.x * coef);
    atomicAdd(ap + 1, v.y * coef);
    atomicAdd(ap + 2, v.z * coef);
    atomicAdd(ap + 3, v.w * coef);
}

// ---------------------------------------------------------------------------
// Finalize: h = [relu]( agg + h * dinv^2 + b )     (in place on h)
// ---------------------------------------------------------------------------
__global__ void gcn_finalize(const float* __restrict__ agg,
                             const float* __restrict__ dinv,
                             const float* __restrict__ b,
                             float* __restrict__ h, int n, int doRelu) {
    long long tid = (long long)blockIdx.x * blockDim.x + threadIdx.x;
    long long node = tid / 24;
    int g = (int)(tid % 24);
    if (node >= n) return;
    float di = dinv[node];
    float sl = di * di;
    float4 hv = *((const float4*)(h + node * HIDC) + g);
    float4 av = *((const float4*)(agg + node * HIDC) + g);
    float4 bv = *((const float4*)b + g);
    float4 r;
    r.x = av.x + hv.x * sl + bv.x;
    r.y = av.y + hv.y * sl + bv.y;
    r.z = av.z + hv.z * sl + bv.z;
    r.w = av.w + hv.w * sl + bv.w;
    if (doRelu) {
        r.x = fmaxf(r.x, 0.0f); r.y = fmaxf(r.y, 0.0f);
        r.z = fmaxf(r.z, 0.0f); r.w = fmaxf(r.w, 0.0f);
    }
    *((float4*)(h + node * HIDC) + g) = r;
}

// ---------------------------------------------------------------------------
// Pool accumulate: pooled[batch[i]] += h[i]; counts[batch[i]] += 1
// ---------------------------------------------------------------------------
__global__ void gcn_pool_accum(const long long* __restrict__ batch,
                               const float* __restrict__ h,
                               float* __restrict__ pooled,
                               float* __restrict__ counts, int n) {
    long long tid = (long long)blockIdx.x * blockDim.x + threadIdx.x;
    long long node = tid / 24;
    int g = (int)(tid % 24);
    if (node >= n) return;
    long long bi = batch[node];
    if (g == 0) atomicAdd(&counts[bi], 1.0f);
    float4 v = *((const float4*)(h + node * HIDC) + g);
    float* p = pooled + bi * HIDC + (long long)g * 4;
    atomicAdd(p + 0, v.x);
    atomicAdd(p + 1, v.y);
    atomicAdd(p + 2, v.z);
    atomicAdd(p + 3, v.w);
}

// out[g,o] = (pooled[g,:] / max(counts[g],1)) @ Wlin[:,o]   (256x16, tiny)
__global__ void gcn_pool_linear(const float* __restrict__ pooled,
                                const float* __restrict__ counts,
                                const float* __restrict__ Wlin,
                                float* __restrict__ out) {
    int tid = blockIdx.x * blockDim.x + threadIdx.x;
    if (tid >= N_GRAPHS * OUTC) return;
    int g = tid / OUTC;
    int o = tid % OUTC;
    float inv = 1.0f / fmaxf(counts[g], 1.0f);
    float s = 0.0f;
    for (int c = 0; c < HIDC; ++c)
        s += pooled[g * HIDC + c] * Wlin[c * OUTC + o];
    out[tid] = s * inv;
}

// ---------------------------------------------------------------------------
extern "C" void kernel_launch(void* const* d_in, const int* in_sizes, int n_in,
                              void* d_out, int out_size, void* d_ws, size_t ws_size,
                              hipStream_t stream) {
    const float*      x     = (const float*)d_in[0];
    const long long*  ei    = (const long long*)d_in[1];
    const long long*  batch = (const long long*)d_in[2];
    const float*      W1    = (const float*)d_in[3];
    const float*      b1    = (const float*)d_in[4];
    const float*      W2    = (const float*)d_in[5];
    const float*      b2    = (const float*)d_in[6];
    const float*      W3    = (const float*)d_in[7];
    const float*      b3    = (const float*)d_in[8];
    const float*      Wlin  = (const float*)d_in[9];
    float*            out   = (float*)d_out;

    // Workspace layout (all 16B aligned): deg/dinv | h | agg | pooled | counts
    float* deg    = (float*)d_ws;                       // N floats (-> dinv)
    float* h      = deg + N_NODES;                      // N*96
    float* agg    = h + (size_t)N_NODES * HIDC;         // N*96
    float* pooled = agg + (size_t)N_NODES * HIDC;       // 256*96
    float* counts = pooled + (size_t)N_GRAPHS * HIDC;   // 256

    const int B = 256;

    // 1) normalization coefficients
    hipMemsetAsync(deg, 0, (size_t)N_NODES * sizeof(float), stream);
    gcn_degree<<<(N_EDGES + B - 1) / B, B, 0, stream>>>(ei, deg, N_EDGES);
    gcn_rsqrt<<<(N_NODES + B - 1) / B, B, 0, stream>>>(deg, N_NODES);

    // 2) three GCN layers
    const float* Ws[3] = {W1, W2, W3};
    const float* bs[3] = {b1, b2, b3};
    const int gemmGrid = (N_NODES + 127) / 128;
    const long long edgeThreads = (long long)N_EDGES * 24;
    const int edgeGrid = (int)((edgeThreads + B - 1) / B);
    const long long nodeThreads = (long long)N_NODES * 24;
    const int nodeGrid = (int)((nodeThreads + B - 1) / B);

    const float* in = x;
    for (int l = 0; l < 3; ++l) {
        gcn_gemm_wmma<<<gemmGrid, B, 0, stream>>>(in, Ws[l], h, N_NODES);
        hipMemsetAsync(agg, 0, (size_t)N_NODES * HIDC * sizeof(float), stream);
        gcn_edge_scatter<<<edgeGrid, B, 0, stream>>>(ei, deg, h, agg, N_EDGES);
        gcn_finalize<<<nodeGrid, B, 0, stream>>>(agg, deg, bs[l], h, N_NODES, l < 2 ? 1 : 0);
        in = h;
    }

    // 3) mean pool + output linear
    hipMemsetAsync(pooled, 0, (size_t)(N_GRAPHS * HIDC + N_GRAPHS) * sizeof(float), stream);
    gcn_pool_accum<<<nodeGrid, B, 0, stream>>>(batch, h, pooled, counts, N_NODES);
    gcn_pool_linear<<<(N_GRAPHS * OUTC + B - 1) / B, B, 0, stream>>>(pooled, counts, Wlin, out);
}